// MedianFilter_37598143709700
// MI455X (gfx1250) — compile-verified
//
#include <hip/hip_runtime.h>
#include <hip/hip_bf16.h>

// -------------------------------------------------------------------------
// 3x3 median filter, fp32, zero 'same' padding, lower median (= true median
// of 9). Shapes fixed by the reference: x is [8, 3, 512, 512] fp32.
//
// MI455X strategy:
//  * Bandwidth-bound (~50 MB -> ~2.2 us at 23.3 TB/s). Stage tiles in LDS
//    via CDNA5 async global->LDS (b128 granules, ASYNCcnt + s_wait_asynccnt)
//    so each element crosses the vector memory path once at 512 B per
//    wave-instruction.
//  * Median via column sort3 + v_med3/v_min3/v_max3; 4 outputs per thread
//    share the 6 column sorts (~13 VALU per output pixel).
// -------------------------------------------------------------------------

#define H 512
#define W 512
#define PLANES 24            // 8 * 3

#define TX 64                // output tile width per block
#define TY 16                // output tile height per block
#define LWV 18               // LDS row stride in float4 granules (72 floats)
#define LW  (LWV * 4)        // 72 floats; rows 288 B = 16-B aligned
#define LH  (TY + 2)         // 18 rows
#define NGRAN (LWV * LH)     // 324 float4 granules per tile
#define XOFF 4               // tile col 0 <-> global col tileX*64 - 4

#define BLOCK 256

#if defined(__HIP_DEVICE_COMPILE__) &&                                        \
    __has_builtin(__builtin_amdgcn_global_load_async_to_lds_b128) &&          \
    __has_builtin(__builtin_amdgcn_s_wait_asynccnt)
#define USE_ASYNC128 1
#else
#define USE_ASYNC128 0
#endif
#if !USE_ASYNC128 && defined(__HIP_DEVICE_COMPILE__) &&                       \
    __has_builtin(__builtin_amdgcn_global_load_async_to_lds_b32) &&           \
    __has_builtin(__builtin_amdgcn_s_wait_asynccnt)
#define USE_ASYNC32 1
#else
#define USE_ASYNC32 0
#endif

#if USE_ASYNC128 || USE_ASYNC32
// Integer round-trip sidesteps C++ addrspace cast rules. AS3 pointers are
// 32-bit; the low 32 bits of a generic LDS pointer are the LDS byte offset.
typedef int v4i_t __attribute__((ext_vector_type(4)));
#define GLOBAL_AS(p)  ((__attribute__((address_space(1))) int*)(unsigned long long)(p))
#define LDS_AS(p)     ((__attribute__((address_space(3))) int*)(unsigned)(unsigned long long)(p))
#define GLOBAL_AS4(p) ((__attribute__((address_space(1))) v4i_t*)(unsigned long long)(p))
#define LDS_AS4(p)    ((__attribute__((address_space(3))) v4i_t*)(unsigned)(unsigned long long)(p))
#endif

__device__ __forceinline__ void sort2(float& a, float& b) {
    float t = fminf(a, b);
    b = fmaxf(a, b);
    a = t;
}

__device__ __forceinline__ void sort3(float& a, float& b, float& c) {
    sort2(a, b);
    sort2(b, c);
    sort2(a, b);
}

__device__ __forceinline__ float med3(float a, float b, float c) {
#if defined(__HIP_DEVICE_COMPILE__)
    return __builtin_amdgcn_fmed3f(a, b, c);   // v_med3_f32 (one VALU op)
#else
    return fmaxf(fminf(a, b), fminf(fmaxf(a, b), c));
#endif
}

__global__ __launch_bounds__(BLOCK) void median3x3_kernel(
    const float* __restrict__ x, float* __restrict__ out) {
    __shared__ float tile[LH * LW];

    const int tid   = threadIdx.x;
    const int plane = blockIdx.z;
    const int gx0   = blockIdx.x * TX - XOFF;  // global col of tile col 0
    const int gy0   = blockIdx.y * TY - 1;     // global row of tile row 0

    const float* src = x + (size_t)plane * (H * W);

    // ---- Fill LDS tile in 16-B granules (zero-padded halo) -------------
    // W == 512 is a multiple of 4, so every granule is either fully inside
    // the image or fully in the pad region: an all-or-nothing test is exact.
    #pragma unroll
    for (int g = tid; g < NGRAN; g += BLOCK) {
        const int lr  = g / LWV;            // tile row 0..17
        const int lgc = g - lr * LWV;       // granule in row 0..17
        const int gr  = gy0 + lr;           // global row
        const int gc  = gx0 + 4 * lgc;      // global col of granule start
        float* ldst = &tile[lr * LW + 4 * lgc];   // 16-B aligned
        if ((unsigned)gr < (unsigned)H && (unsigned)gc < (unsigned)W) {
            const float* gsrc = src + (size_t)gr * W + gc;  // 16-B aligned
#if USE_ASYNC128
            // Per-lane DMA: 16 B memory -> LDS, no VGPR data, ASYNCcnt.
            __builtin_amdgcn_global_load_async_to_lds_b128(
                GLOBAL_AS4(gsrc), LDS_AS4(ldst), 0, 0);
#elif USE_ASYNC32
            __builtin_amdgcn_global_load_async_to_lds_b32(GLOBAL_AS(gsrc), LDS_AS(ldst), 0, 0);
            __builtin_amdgcn_global_load_async_to_lds_b32(GLOBAL_AS(gsrc), LDS_AS(ldst), 4, 4);
            __builtin_amdgcn_global_load_async_to_lds_b32(GLOBAL_AS(gsrc), LDS_AS(ldst), 8, 8);
            __builtin_amdgcn_global_load_async_to_lds_b32(GLOBAL_AS(gsrc), LDS_AS(ldst), 12, 12);
#else
            *(float4*)ldst = *(const float4*)gsrc;
#endif
        } else {
            float4 z; z.x = 0.f; z.y = 0.f; z.z = 0.f; z.w = 0.f;
            *(float4*)ldst = z;             // zero padding (ds_store_b128)
        }
    }
#if USE_ASYNC128 || USE_ASYNC32
    __builtin_amdgcn_s_wait_asynccnt(0);
#endif
    __syncthreads();

    // ---- Compute 4 horizontally adjacent medians per thread ------------
    const int txg = tid & 15;   // x-group: 4 outputs each -> 16 groups span TX
    const int ty  = tid >> 4;   // 0..15 output row inside tile

    // output col o needs tile cols o+3 .. o+5 (tile col t <-> global col
    // gx0 + t = tileX*64 + t - 4); this thread: cols 4*txg+3 .. 4*txg+8.
    const int cbase = 4 * txg + (XOFF - 1);

    float lo[6], mi[6], hi[6];
    #pragma unroll
    for (int j = 0; j < 6; ++j) {
        float a = tile[(ty + 0) * LW + cbase + j];
        float b = tile[(ty + 1) * LW + cbase + j];
        float c = tile[(ty + 2) * LW + cbase + j];
        sort3(a, b, c);                 // column-wise min / med / max
        lo[j] = a; mi[j] = b; hi[j] = c;
    }

    float4 r;
    float* rp = &r.x;
    #pragma unroll
    for (int k = 0; k < 4; ++k) {
        const float mxlo = fmaxf(fmaxf(lo[k], lo[k + 1]), lo[k + 2]); // v_max3
        const float mdmd = med3(mi[k], mi[k + 1], mi[k + 2]);         // v_med3
        const float mnhi = fminf(fminf(hi[k], hi[k + 1]), hi[k + 2]); // v_min3
        rp[k] = med3(mxlo, mdmd, mnhi); // median of the 3 surviving candidates
    }

    const int orow = blockIdx.y * TY + ty;
    const int ocol = blockIdx.x * TX + 4 * txg;          // 16-B aligned
    *(float4*)(out + ((size_t)plane * H + orow) * W + ocol) = r;  // b128 store
}

extern "C" void kernel_launch(void* const* d_in, const int* in_sizes, int n_in,
                              void* d_out, int out_size, void* d_ws, size_t ws_size,
                              hipStream_t stream) {
    (void)in_sizes; (void)n_in; (void)d_ws; (void)ws_size; (void)out_size;
    const float* x = (const float*)d_in[0];
    float* out = (float*)d_out;

    dim3 grid(W / TX, H / TY, PLANES);   // (8, 32, 24)
    dim3 block(BLOCK);
    median3x3_kernel<<<grid, block, 0, stream>>>(x, out);
}